// LightGCN_26216480375154
// MI455X (gfx1250) — compile-verified
//
#include <hip/hip_runtime.h>

// ---------------- problem constants (match reference) ----------------
#define NUM_USERS 200000
#define NUM_ITEMS 100000
#define NNODES    (NUM_USERS + NUM_ITEMS)
#define DIM       32
#define NLAYERS   3

// ---------------- SpMM tiling ----------------
#define CHUNK            1024                 // edges staged per block-iteration
#define SPMM_BLOCK       256                  // 8 wave32s
#define WAVES_PER_BLOCK  (SPMM_BLOCK / 32)
#define EDGES_PER_WAVE   (CHUNK / WAVES_PER_BLOCK)   // 128

typedef int v4i __attribute__((ext_vector_type(4)));

// ---------------- CDNA5 async global->LDS copy (ASYNCcnt path) ----------------
// gfx1250: GLOBAL_LOAD_ASYNC_TO_LDS_B128, tracked with ASYNCcnt (ISA 08 §4).
// Builtin signature: (v4i AS1* src, v4i AS3* dst, imm offset, imm cpol).
__device__ __forceinline__ void async_copy_b128(const void* g, void* l) {
#if __has_builtin(__builtin_amdgcn_global_load_async_to_lds_b128)
  __builtin_amdgcn_global_load_async_to_lds_b128(
      (__attribute__((address_space(1))) v4i*)g,
      (__attribute__((address_space(3))) v4i*)l,
      /*offset=*/0, /*cpol=*/0);
#else
  *(int4*)l = *(const int4*)g;
#endif
}

#if __has_builtin(__builtin_amdgcn_s_wait_asynccnt)
#define WAIT_ASYNC_LE(n) __builtin_amdgcn_s_wait_asynccnt(n)
#else
#define WAIT_ASYNC_LE(n) asm volatile("s_wait_asynccnt %0" ::"i"(n) : "memory")
#endif

// ---------------- SpMM: y[row] += val * x[col], wave-per-edge, lane = feature dim ----------------
__global__ __launch_bounds__(SPMM_BLOCK) void spmm_atomic_kernel(
    const int* __restrict__ rows, const int* __restrict__ cols,
    const float* __restrict__ vals, const float* __restrict__ x,
    float* __restrict__ y, int nEdges) {
  __shared__ int   s_rows[2][CHUNK];
  __shared__ int   s_cols[2][CHUNK];
  __shared__ float s_vals[2][CHUNK];

  const int tid  = threadIdx.x;
  const int lane = tid & 31;
  const int wave = tid >> 5;
  const int nChunks = (nEdges + CHUNK - 1) / CHUNK;

  // Stage one chunk: each of 256 threads async-copies 16B per array (3x b128).
  // Returns true iff the async path was used (full chunk).
  auto issue = [&](int c, int b) -> bool {
    const long base = (long)c * CHUNK;
    const int  off  = tid * 4;  // 4 ints/floats per thread -> 1024 per block
    if (base + CHUNK <= (long)nEdges) {
      async_copy_b128(rows + base + off, &s_rows[b][off]);
      async_copy_b128(cols + base + off, &s_cols[b][off]);
      async_copy_b128(vals + base + off, &s_vals[b][off]);
      return true;
    }
    // guarded tail (synchronous; visibility handled by __syncthreads)
    for (int k = 0; k < 4; ++k) {
      const long e = base + off + k;
      if (e < (long)nEdges) {
        s_rows[b][off + k] = rows[e];
        s_cols[b][off + k] = cols[e];
        s_vals[b][off + k] = vals[e];
      }
    }
    return false;
  };

  int c = blockIdx.x;
  if (c >= nChunks) return;  // uniform per block
  int buf = 0;
  issue(c, 0);

  for (; c < nChunks; c += gridDim.x) {
    const int nxt = c + gridDim.x;
    bool nxtAsync = false;
    if (nxt < nChunks) nxtAsync = issue(nxt, buf ^ 1);  // prefetch next chunk
    // Exact drain: if the next chunk went out async, 3 of its ops are pending and
    // waiting to <=3 retires chunk c's copies (async loads complete in order).
    // Otherwise drain fully so chunk c's copies are guaranteed resident.
    if (nxtAsync) { WAIT_ASYNC_LE(3); } else { WAIT_ASYNC_LE(0); }
    __syncthreads();         // all waves' copies of chunk c visible in LDS

    const long base = (long)c * CHUNK;
    const int  jBeg = wave * EDGES_PER_WAVE;
    const long rem  = (long)nEdges - base - jBeg;
    const int  jCnt = rem >= EDGES_PER_WAVE ? EDGES_PER_WAVE
                                            : (rem > 0 ? (int)rem : 0);
#pragma unroll 4
    for (int j = 0; j < jCnt; ++j) {
      const int   r  = s_rows[buf][jBeg + j];
      const int   cc = s_cols[buf][jBeg + j];
      const float v  = s_vals[buf][jBeg + j];
      if (j + 16 < jCnt) {   // software prefetch of gather stream (global_prefetch_b8)
        __builtin_prefetch(&x[(long)s_cols[buf][jBeg + j + 16] * DIM + lane], 0, 0);
      }
      const float xv = x[(long)cc * DIM + lane];        // 128B coalesced gather (L2-resident)
      unsafeAtomicAdd(&y[(long)r * DIM + lane], v * xv); // no-return global_atomic_add_f32
    }
    __syncthreads();         // all waves done reading buf before it is refilled
    buf ^= 1;
  }
}

// ---------------- elementwise helpers ----------------
__global__ void init_x_kernel(const float* __restrict__ uw, const float* __restrict__ iw,
                              float* __restrict__ xcur, float* __restrict__ acc, long n) {
  const long uElems = (long)NUM_USERS * DIM;
  long i = (long)blockIdx.x * blockDim.x + threadIdx.x;
  const long stride = (long)gridDim.x * blockDim.x;
  for (; i < n; i += stride) {
    const float v = (i < uElems) ? uw[i] : iw[i - uElems];
    xcur[i] = v;
    acc[i]  = v;
  }
}

__global__ void zero_kernel(float* __restrict__ p, long n) {
  long i = (long)blockIdx.x * blockDim.x + threadIdx.x;
  const long stride = (long)gridDim.x * blockDim.x;
  for (; i < n; i += stride) p[i] = 0.0f;
}

__global__ void add_scale_kernel(float* __restrict__ acc, const float* __restrict__ xn,
                                 float s, long n) {
  long i = (long)blockIdx.x * blockDim.x + threadIdx.x;
  const long stride = (long)gridDim.x * blockDim.x;
  for (; i < n; i += stride) acc[i] = (acc[i] + xn[i]) * s;
}

// ---------------- host orchestration ----------------
extern "C" void kernel_launch(void* const* d_in, const int* in_sizes, int n_in,
                              void* d_out, int out_size, void* d_ws, size_t ws_size,
                              hipStream_t stream) {
  const int*   rows = (const int*)d_in[0];
  const int*   cols = (const int*)d_in[1];
  const float* vals = (const float*)d_in[2];
  const float* uw   = (const float*)d_in[3];
  const float* iw   = (const float*)d_in[4];
  const int    E    = in_sizes[0];

  float* out = (float*)d_out;
  float* xa  = (float*)d_ws;                       // x ping
  float* xb  = xa + (long)NNODES * DIM;            // x pong

  const long nFeat = (long)NNODES * DIM;           // 9.6M floats
  const int  EW_GRID = 4096, EW_BLOCK = 256;

  init_x_kernel<<<EW_GRID, EW_BLOCK, 0, stream>>>(uw, iw, xa, out, nFeat);

  const int nChunks = (E + CHUNK - 1) / CHUNK;
  const int sGrid   = nChunks < 3072 ? nChunks : 3072;

  float* cur = xa;
  float* nxt = xb;
  for (int l = 0; l < NLAYERS; ++l) {
    zero_kernel<<<EW_GRID, EW_BLOCK, 0, stream>>>(nxt, nFeat);
    spmm_atomic_kernel<<<sGrid, SPMM_BLOCK, 0, stream>>>(rows, cols, vals, cur, nxt, E);
    const float s = (l == NLAYERS - 1) ? (1.0f / (NLAYERS + 1)) : 1.0f;
    add_scale_kernel<<<EW_GRID, EW_BLOCK, 0, stream>>>(out, nxt, s, nFeat);
    float* t = cur; cur = nxt; nxt = t;            // ping-pong
  }
}